// ModulatedMultiHeadAttention_38019050504406
// MI455X (gfx1250) — compile-verified
//
#include <hip/hip_runtime.h>

// ---------------------------------------------------------------------------
// ModulatedMultiHeadAttention for MI455X (gfx1250, wave32, WMMA)
// B=4, N=8193 (1 cls + 8192 tokens), C=768, H=1, D=768, TOPK=12
// ---------------------------------------------------------------------------

typedef float v2f __attribute__((ext_vector_type(2)));
typedef float v8f __attribute__((ext_vector_type(8)));

#define TM 128
#define TN 128
#define TK 16
#define LPAD 20     // LDS row stride (floats): n*20 mod 64 distinct -> conflict-free

// Async global->LDS 16-byte copy (CDNA5 GLOBAL_LOAD_ASYNC_TO_LDS_B128).
// Tracked by ASYNCcnt; completion enforced with s_wait_asynccnt before barrier.
__device__ __forceinline__ void async_copy16(const float* gsrc, float* lds_dst) {
  uint32_t loff = (uint32_t)(uintptr_t)lds_dst;   // low 32 bits = LDS offset
  asm volatile("global_load_async_to_lds_b128 %0, %1, off"
               :: "v"(loff), "v"(gsrc) : "memory");
}
__device__ __forceinline__ void wait_async0() {
  asm volatile("s_wait_asynccnt 0x0" ::: "memory");
}

// ============================================================================
// Kernel 1: qkv = x @ W^T + b   (M x 2304, K = 768) via V_WMMA_F32_16X16X4_F32
// Both A (x rows) and B (W rows) are staged untransposed: the WMMA B operand
// per lane is two consecutive k of one W row, so Bs[n][k] fragments are a
// single aligned 8-byte ds load per WMMA (no transpose, no movs).
// ============================================================================
__global__ __launch_bounds__(256) void qkv_gemm_wmma(
    const float* __restrict__ x, const float* __restrict__ W,
    const float* __restrict__ bias, float* __restrict__ qkv, int M)
{
  __shared__ float As[2][TM][LPAD];   // x tile:  128 rows x 16 k
  __shared__ float Bs[2][TN][LPAD];   // W tile:  128 rows x 16 k (untransposed)

  const int tid  = threadIdx.x;
  const int wave = tid >> 5;          // 0..7, each wave: 16 rows x 128 cols
  const int lane = tid & 31;
  const int ml   = lane & 15;         // M pos (A frag) / N pos (B, C frag)
  const int kh2  = (lane >> 4) << 1;  // 0 or 2: K-pair selector per ISA layout
  const int m0   = blockIdx.x * TM;
  const int n0   = blockIdx.y * TN;

  v8f acc[8] = {};                    // 8 tiles of 16x16 f32 accumulators

  // Stage one 128x16 A tile + 128x16 B tile: 256 threads x 2 chunks x 16B each
  auto stage = [&](int buf, int k0) {
    #pragma unroll
    for (int h = 0; h < 2; ++h) {
      int chunk = tid + h * 256;          // 0..511
      int r = chunk >> 2;                 // row 0..127
      int q = (chunk & 3) * 4;            // 4-float quarter of the row
      int mr = m0 + r; if (mr >= M) mr = M - 1;   // clamp (stores are guarded)
      async_copy16(x + (size_t)mr * 768 + k0 + q,        &As[buf][r][q]);
      async_copy16(W + (size_t)(n0 + r) * 768 + k0 + q,  &Bs[buf][r][q]);
    }
  };

  stage(0, 0);
  wait_async0();
  __syncthreads();

  const int rA = wave * 16;
  for (int kt = 0; kt < 48; ++kt) {                 // 768 / 16
    int cur = kt & 1;
    if (kt + 1 < 48) stage(cur ^ 1, (kt + 1) * TK); // async fill of next buffer
    #pragma unroll
    for (int ks = 0; ks < 4; ++ks) {                // 4 WMMA K=4 steps / chunk
      const int kk = ks * 4 + kh2;
      // A 16x4 frag: VGPR0 = K{0,2}, VGPR1 = K{1,3}: contiguous pair per lane
      v2f a2 = *(const v2f*)&As[cur][rA + ml][kk];
      #pragma unroll
      for (int nt = 0; nt < 8; ++nt) {
        // B 4x16 frag: lane ml holds (K=kh2, n) and (K=kh2+1, n): one b64 load
        v2f b2 = *(const v2f*)&Bs[cur][nt * 16 + ml][kk];
        acc[nt] = __builtin_amdgcn_wmma_f32_16x16x4_f32(
            false, a2, false, b2, (short)0, acc[nt], false, false);
      }
    }
    wait_async0();                                  // next-buffer copies landed
    __syncthreads();
  }

  // Epilogue: C/D layout -> VGPR i: lanes0-15 M=i, lanes16-31 M=8+i; N=ml
  #pragma unroll
  for (int nt = 0; nt < 8; ++nt) {
    int n = n0 + nt * 16 + ml;
    float bv = bias[n];
    #pragma unroll
    for (int i = 0; i < 8; ++i) {
      int m = m0 + rA + (kh2 << 2) + i;             // kh2<<2 == (lane>>4)*8
      if (m < M) qkv[(size_t)m * 2304 + n] = acc[nt][i] + bv;
    }
  }
}

// ============================================================================
// Top-12 register-resident sorted insert (descending, stable on ties)
// ============================================================================
__device__ __forceinline__ void topk_insert(float (&vals)[12], float v) {
  if (v > vals[11]) {
    vals[11] = v;
    #pragma unroll
    for (int j = 11; j >= 1; --j) {
      bool sw = vals[j] > vals[j - 1];
      float a = vals[j - 1], b = vals[j];
      vals[j - 1] = sw ? b : a;
      vals[j]     = sw ? a : b;
    }
  }
}
__device__ __forceinline__ void topk_insert_idx(float (&vals)[12], int (&idx)[12],
                                                float v, int t) {
  if (v > vals[11]) {
    vals[11] = v; idx[11] = t;
    #pragma unroll
    for (int j = 11; j >= 1; --j) {
      bool sw = vals[j] > vals[j - 1];
      float av = vals[j - 1], bv = vals[j];
      int   ai = idx[j - 1],  bi = idx[j];
      vals[j - 1] = sw ? bv : av; idx[j - 1] = sw ? bi : ai;
      vals[j]     = sw ? av : bv; idx[j]     = sw ? ai : bi;
    }
  }
}

// ============================================================================
// Kernel 2: modulation + mod_v write + chunked partial top-12
// grid: (3 channel groups, 32 token chunks, 4 batches), block 256
// ============================================================================
__global__ __launch_bounds__(256) void modv_topk(
    const float* __restrict__ qkv, float* __restrict__ out,
    float* __restrict__ pQ, float* __restrict__ pK,
    float* __restrict__ pV, int* __restrict__ pVi)
{
  const int d     = blockIdx.x * 256 + threadIdx.x;  // 0..767
  const int chunk = blockIdx.y;                      // 0..31
  const int b     = blockIdx.z;                      // 0..3
  const size_t base = (size_t)b * 8193 * 2304;

  const float qc = qkv[base + d];
  const float kc = qkv[base + 768 + d];
  const float vc = qkv[base + 1536 + d];
  const float iv = qc * kc * (1.0f + fabsf(vc));

  float lq[12], lk[12], lv[12]; int li[12];
  #pragma unroll
  for (int j = 0; j < 12; ++j) {
    lq[j] = -__builtin_inff(); lk[j] = -__builtin_inff();
    lv[j] = -__builtin_inff(); li[j] = 0;
  }

  const int t0 = chunk * 256;
  for (int t = t0; t < t0 + 256; ++t) {
    size_t row = base + (size_t)(1 + t) * 2304;
    __builtin_prefetch(&qkv[row + (size_t)8 * 2304 + d], 0, 1); // global_prefetch_b8
    float q = qkv[row + d];
    float k = qkv[row + 768 + d];
    float v = qkv[row + 1536 + d];
    float aq = qc * k + q;               // cls_q * k_p + q_p
    float ak = kc * q + k;               // cls_k * q_p + k_p
    float av = v * v + 2.0f * v + iv;    // v_p^2 + 2 v_p + inter_v
    out[((size_t)b * 8193 + 1 + t) * 768 + d] = av;   // mod_v into output
    topk_insert(lq, aq);
    topk_insert(lk, ak);
    topk_insert_idx(lv, li, av, t);
  }

  size_t pbase = (((size_t)b * 768 + d) * 32 + chunk) * 12;
  #pragma unroll
  for (int j = 0; j < 12; ++j) {
    pQ[pbase + j]  = lq[j];
    pK[pbase + j]  = lk[j];
    pV[pbase + j]  = lv[j];
    pVi[pbase + j] = li[j];
  }
}

// ============================================================================
// Kernel 3: merge 32 sorted partial lists -> global top-12 per (b, d, stream)
// ============================================================================
__device__ __forceinline__ void merge32(const float* __restrict__ src,
                                        const int* __restrict__ srci,
                                        float* __restrict__ dstv,
                                        int* __restrict__ dsti,
                                        int b, int d) {
  int ptr[32];
  for (int c = 0; c < 32; ++c) ptr[c] = 0;
  for (int j = 0; j < 12; ++j) {
    float best = -__builtin_inff(); int bc = 0;
    for (int c = 0; c < 32; ++c) {
      if (ptr[c] < 12) {
        float v = src[(size_t)c * 12 + ptr[c]];
        if (v > best) { best = v; bc = c; }       // ascending c => lowest idx wins ties
      }
    }
    dstv[((size_t)b * 12 + j) * 768 + d] = best;
    if (dsti) dsti[((size_t)b * 12 + j) * 768 + d] = srci[(size_t)bc * 12 + ptr[bc]];
    ptr[bc]++;
  }
}

__global__ __launch_bounds__(256) void merge_topk(
    const float* __restrict__ pQ, const float* __restrict__ pK,
    const float* __restrict__ pV, const int* __restrict__ pVi,
    float* __restrict__ mQ, float* __restrict__ mK,
    float* __restrict__ mV, int* __restrict__ mVi)
{
  int tid = blockIdx.x * blockDim.x + threadIdx.x;
  if (tid >= 4 * 768) return;
  int b = tid / 768, d = tid % 768;
  size_t pbase = ((size_t)b * 768 + d) * 32 * 12;
  merge32(pQ + pbase, nullptr, mQ, nullptr, b, d);
  merge32(pK + pbase, nullptr, mK, nullptr, b, d);
  merge32(pV + pbase, pVi + pbase, mV, mVi, b, d);
}

// ============================================================================
// Kernel 4: 13x13 attention + cls row + scatter-add of out_rest into mod_v
// grid: 4 (one block per batch), block 256
// ============================================================================
__global__ __launch_bounds__(256) void attn_final(
    const float* __restrict__ qkv, const float* __restrict__ mQ,
    const float* __restrict__ mK, const float* __restrict__ mV,
    const int* __restrict__ mVi, float* __restrict__ out)
{
  const int b   = blockIdx.x;
  const int tid = threadIdx.x;
  __shared__ float P[13][13];
  const float scale = 0.036084391824351615f;  // 768^-0.5
  const float* cls = qkv + (size_t)b * 8193 * 2304;

  // S[i][j] = qm[i,:] . km[j,:]
  for (int p = tid; p < 169; p += 256) {
    int i = p / 13, j = p % 13;
    const float* qrow = (i == 0) ? cls        : (mQ + ((size_t)b * 12 + i - 1) * 768);
    const float* krow = (j == 0) ? cls + 768  : (mK + ((size_t)b * 12 + j - 1) * 768);
    float s = 0.0f;
    for (int d = 0; d < 768; ++d) s += qrow[d] * krow[d];
    P[i][j] = s * scale;
  }
  __syncthreads();

  if (tid < 13) {
    float mx = -__builtin_inff();
    for (int j = 0; j < 13; ++j) mx = fmaxf(mx, P[tid][j]);
    float sum = 0.0f;
    for (int j = 0; j < 13; ++j) { float e = __expf(P[tid][j] - mx); P[tid][j] = e; sum += e; }
    float inv = 1.0f / sum;
    for (int j = 0; j < 13; ++j) P[tid][j] *= inv;
  }
  __syncthreads();

  // out[i,:] = P[i,:] @ vm ; row 0 -> cls output row, rows 1..12 scatter-add
  for (int p = tid; p < 13 * 768; p += 256) {
    int i = p / 768, d = p % 768;
    float o = 0.0f;
    for (int j = 0; j < 13; ++j) {
      float vv = (j == 0) ? cls[1536 + d] : mV[((size_t)b * 12 + j - 1) * 768 + d];
      o += P[i][j] * vv;
    }
    if (i == 0) {
      out[(size_t)b * 8193 * 768 + d] = o;
    } else {
      int t = mVi[((size_t)b * 12 + i - 1) * 768 + d];  // distinct per (b,d): no race
      size_t addr = ((size_t)b * 8193 + 1 + t) * 768 + d;
      out[addr] += o;                                   // mod_v already there
    }
  }
}

// ============================================================================
extern "C" void kernel_launch(void* const* d_in, const int* in_sizes, int n_in,
                              void* d_out, int out_size, void* d_ws, size_t ws_size,
                              hipStream_t stream) {
  const float* x    = (const float*)d_in[0];   // (4, 8193, 768)
  const float* W    = (const float*)d_in[1];   // (2304, 768)
  const float* bias = (const float*)d_in[2];   // (2304,)
  float* out = (float*)d_out;                  // (4, 8193, 768)
  float* ws  = (float*)d_ws;

  const int M = 4 * 8193;                      // 32772 rows
  const size_t QKV  = (size_t)M * 2304;        // 75,506,688 floats
  const size_t PART = (size_t)4 * 768 * 32 * 12;
  const size_t MER  = (size_t)4 * 12 * 768;

  float* qkv = ws;
  float* pQ  = ws + QKV;
  float* pK  = pQ + PART;
  float* pV  = pK + PART;
  int*   pVi = (int*)(pV + PART);
  float* mQ  = (float*)(pVi + PART);
  float* mK  = mQ + MER;
  float* mV  = mK + MER;
  int*   mVi = (int*)(mV + MER);

  dim3 gA((M + TM - 1) / TM, 2304 / TN);       // 257 x 18 workgroups
  qkv_gemm_wmma<<<gA, 256, 0, stream>>>(x, W, bias, qkv, M);

  dim3 gB(3, 32, 4);                           // channel groups x chunks x batch
  modv_topk<<<gB, 256, 0, stream>>>(qkv, out, pQ, pK, pV, pVi);

  merge_topk<<<12, 256, 0, stream>>>(pQ, pK, pV, pVi, mQ, mK, mV, mVi);

  attn_final<<<4, 256, 0, stream>>>(qkv, mQ, mK, mV, mVi, out);
}